// LM_MLE_simflux_70171175682102
// MI455X (gfx1250) — compile-verified
//
#include <hip/hip_runtime.h>
#include <stdint.h>

// LM-MLE simflux update, MI455X (gfx1250, wave32).
// Bandwidth-bound streaming reduction (~352 MB @ 23.3 TB/s ~= 15 us floor).
// CDNA5 paths: global_load_async_to_lds_b128 double-buffered jacobian staging
// (ASYNCcnt), s_wait_asynccnt / s_wait_dscnt split counters, packed-f32 math,
// and a V_WMMA_F32_16X16X4_F32 ones-matrix cross-lane reduction (C[m][n] =
// sum_k B[k][n] reduces 4 lane-slots per column per op; 8 chained WMMAs
// reduce all 32 lanes for 16 accumulators -> cheaper than a shuffle tree).

#define LAMBDA_C 0.01f
#define KEPS_C   2.220446049250313e-13f   // 1000 * FLT64_EPS, as in reference
#define LN2F     0.69314718055994530942f

#define WAVES_PB   8
#define THREADS_PB 256
#define CHUNK      64        // spatial samples per wave per iteration
#define PDIM       5
#define RSTRIDE    24        // padded row stride (floats) for reduction matrix

typedef float v2f __attribute__((ext_vector_type(2)));
typedef float v8f __attribute__((ext_vector_type(8)));

__device__ __forceinline__ float2 f2add(float2 a, float2 b){ return make_float2(a.x+b.x, a.y+b.y); }
__device__ __forceinline__ float2 f2sub(float2 a, float2 b){ return make_float2(a.x-b.x, a.y-b.y); }
__device__ __forceinline__ float2 f2mul(float2 a, float2 b){ return make_float2(a.x*b.x, a.y*b.y); }
__device__ __forceinline__ float2 f2fma(float2 a, float2 b, float2 c){
  return make_float2(__builtin_fmaf(a.x,b.x,c.x), __builtin_fmaf(a.y,b.y,c.y));
}

// Copy 1280 contiguous bytes global -> LDS with the CDNA5 async DMA path.
// 80 x 16B lane-transfers = 3 instructions (3rd uses lanes 0..15 only).
// The 24-bit IOFFSET applies to BOTH the LDS and global addresses.
__device__ __forceinline__ void async_copy_1280(unsigned lds_off, const float* gsrc, int lane){
  uint64_t ga = (uint64_t)(uintptr_t)gsrc + (unsigned)(lane * 16);
  unsigned  la = lds_off + (unsigned)(lane * 16);
  asm volatile("global_load_async_to_lds_b128 %0, %1, off"             :: "v"(la), "v"(ga) : "memory");
  asm volatile("global_load_async_to_lds_b128 %0, %1, off offset:512"  :: "v"(la), "v"(ga) : "memory");
  if (lane < 16)
    asm volatile("global_load_async_to_lds_b128 %0, %1, off offset:1024" :: "v"(la), "v"(ga) : "memory");
}

__global__ __launch_bounds__(THREADS_PB)
void lm_mle_simflux_kernel(const float* __restrict__ smp,
                           const float* __restrict__ mu,
                           const float* __restrict__ jac,
                           const float* __restrict__ cur,
                           const float* __restrict__ prmm,
                           float* __restrict__ out,
                           int Btot, int Nsp)
{
  // per-wave DMA double buffer: 8 waves * 2 * 1280B = 20.5 KB
  __shared__ __align__(16) float jbuf[WAVES_PB][2][CHUNK * PDIM];
  // per-wave 32x21 partials matrix (padded to 24 cols) + overrun pad
  __shared__ __align__(16) float rbuf[WAVES_PB * 32 * RSTRIDE + 32];

  const int lane = (int)(threadIdx.x & 31u);
  const int wid  = (int)(threadIdx.x >> 5);
  const int b    = blockIdx.x * WAVES_PB + wid;
  if (b >= Btot) return;                       // wave-uniform exit

  const size_t baseN = (size_t)b * (size_t)Nsp;
  const float* jg = jac + baseN * PDIM;
  const float* sg = smp + baseN;
  const float* mg = mu  + baseN;
  const int niter = Nsp / CHUNK;               // 1536/64 = 24

  unsigned lbase0 = (unsigned)(uintptr_t)(&jbuf[wid][0][0]);
  unsigned lbase1 = (unsigned)(uintptr_t)(&jbuf[wid][1][0]);

  float2 hacc[15], gacc[5], lacc;
  #pragma unroll
  for (int i = 0; i < 15; ++i) hacc[i] = make_float2(0.f, 0.f);
  #pragma unroll
  for (int i = 0; i < 5;  ++i) gacc[i] = make_float2(0.f, 0.f);
  lacc = make_float2(0.f, 0.f);

  // prologue: DMA chunk 0
  async_copy_1280(lbase0, jg, lane);

  for (int it = 0; it < niter; ++it) {
    if (it + 1 < niter) {
      // target buffer of next DMA == buffer read in the previous iteration:
      // make sure those LDS reads have drained before the engine overwrites it.
      asm volatile("s_wait_dscnt 0" ::: "memory");
      async_copy_1280(((it + 1) & 1) ? lbase1 : lbase0,
                      jg + (size_t)(it + 1) * (CHUNK * PDIM), lane);
      // async loads complete in order: <=3 outstanding => current chunk landed
      asm volatile("s_wait_asynccnt 3" ::: "memory");
    } else {
      asm volatile("s_wait_asynccnt 0" ::: "memory");
    }

    const int n0 = it * CHUNK + 2 * lane;          // two samples per lane
    float2 sp = *(const float2*)(sg + n0);         // coalesced b64
    float2 mv = *(const float2*)(mg + n0);

    // 40B/lane from LDS, stride 10 dwords -> conflict-free on 64 banks
    const float* jl = &jbuf[wid][it & 1][lane * 10];
    float2 c0 = ((const float2*)jl)[0];
    float2 c1 = ((const float2*)jl)[1];
    float2 c2 = ((const float2*)jl)[2];
    float2 c3 = ((const float2*)jl)[3];
    float2 c4 = ((const float2*)jl)[4];
    float2 jv[5];
    jv[0] = make_float2(c0.x, c2.y);
    jv[1] = make_float2(c0.y, c3.x);
    jv[2] = make_float2(c1.x, c3.y);
    jv[3] = make_float2(c1.y, c4.x);
    jv[4] = make_float2(c2.x, c4.y);

    float2 mp;
    mp.x = mv.x > 0.f ? mv.x : (mv.x < 0.f ? KEPS_C : 0.f);
    mp.y = mv.y > 0.f ? mv.y : (mv.y < 0.f ? KEPS_C : 0.f);
    float2 r  = make_float2(__builtin_amdgcn_rcpf(mp.x), __builtin_amdgcn_rcpf(mp.y));
    float2 lg = make_float2(__builtin_amdgcn_logf(mp.x) * LN2F,
                            __builtin_amdgcn_logf(mp.y) * LN2F);
    float2 w  = f2mul(f2sub(sp, mp), r);           // (smp-mu)/mu
    float2 dw = f2mul(f2mul(sp, r), r);            // smp/mu^2
    lacc = f2add(lacc, f2sub(f2mul(sp, lg), mp));  // smp*log(mu) - mu

    float2 t[5];
    #pragma unroll
    for (int p = 0; p < 5; ++p) {
      gacc[p] = f2fma(w, jv[p], gacc[p]);
      t[p] = f2mul(dw, jv[p]);
    }
    int c = 0;
    #pragma unroll
    for (int p = 0; p < 5; ++p)
      #pragma unroll
      for (int q = p; q < 5; ++q) { hacc[c] = f2fma(t[p], jv[q], hacc[c]); ++c; }
  }

  // ---- cross-lane reduction of 21 partials via WMMA ones-matrix trick ----
  // Scatter partials to LDS as M[srcLane][acc], then C = ones(16x4) x B
  // chained over 8 K-chunks sums all 32 lanes per accumulator column.
  float part[21];
  #pragma unroll
  for (int i = 0; i < 15; ++i) part[i] = hacc[i].x + hacc[i].y;
  #pragma unroll
  for (int i = 0; i < 5;  ++i) part[15 + i] = gacc[i].x + gacc[i].y;
  part[20] = lacc.x + lacc.y;

  float* rb = &rbuf[wid * 32 * RSTRIDE];
  {
    float* wr = rb + lane * RSTRIDE;            // 96B row stride, 16B aligned
    *(float4*)(wr +  0) = make_float4(part[0],  part[1],  part[2],  part[3]);
    *(float4*)(wr +  4) = make_float4(part[4],  part[5],  part[6],  part[7]);
    *(float4*)(wr +  8) = make_float4(part[8],  part[9],  part[10], part[11]);
    *(float4*)(wr + 12) = make_float4(part[12], part[13], part[14], part[15]);
    *(float4*)(wr + 16) = make_float4(part[16], part[17], part[18], part[19]);
    wr[20] = part[20];
  }
  asm volatile("" ::: "memory");   // keep stores before loads (HW keeps DS in order)

  v2f ones; ones[0] = 1.0f; ones[1] = 1.0f;
  const int half = lane >> 4;                   // 0: k-slots {0,1}, 1: {2,3}
  const int col  = lane & 15;
  float sums0 = 0.0f, sums1 = 0.0f;
  #pragma unroll
  for (int g = 0; g < 2; ++g) {
    v8f acc = {0.f, 0.f, 0.f, 0.f, 0.f, 0.f, 0.f, 0.f};
    #pragma unroll
    for (int j = 0; j < 8; ++j) {               // 4 source lanes per WMMA
      const int r0 = 4 * j + 2 * half;
      v2f bv;
      bv[0] = rb[r0       * RSTRIDE + g * 16 + col];
      bv[1] = rb[(r0 + 1) * RSTRIDE + g * 16 + col];
      acc = __builtin_amdgcn_wmma_f32_16x16x4_f32(false, ones, false, bv,
                                                  (short)0, acc, false, false);
    }
    // all rows of C equal the column sums; C[0][col] sits in acc[0]
    if (g == 0) sums0 = acc[0]; else sums1 = acc[0];
  }

  float red[21];
  #pragma unroll
  for (int i = 0; i < 16; ++i) red[i] = __shfl(sums0, i, 32);
  #pragma unroll
  for (int i = 16; i < 21; ++i) red[i] = __shfl(sums1, i - 16, 32);

  if (lane == 0) {
    // -Bmat = hss + lambda*diag(hss);  solve (-Bmat) x = grad
    float A[5][5], x[5];
    #pragma unroll
    for (int p = 0; p < 5; ++p) {
      #pragma unroll
      for (int q = 0; q < 5; ++q) {
        int i = (p <= q) ? (p*5 - p*(p+1)/2 + q) : (q*5 - q*(q+1)/2 + p);
        A[p][q] = red[i];
      }
      A[p][p] *= (1.0f + LAMBDA_C);
      x[p] = red[15 + p];
    }
    // unrolled no-pivot Gaussian elimination (A is SPD after LM damping);
    // a singular system yields inf/NaN -> handled by the NaN fallback below.
    #pragma unroll
    for (int k = 0; k < 5; ++k) {
      float inv = 1.0f / A[k][k];
      #pragma unroll
      for (int ri = k + 1; ri < 5; ++ri) {
        float f = A[ri][k] * inv;
        #pragma unroll
        for (int q = k; q < 5; ++q) A[ri][q] = __builtin_fmaf(-f, A[k][q], A[ri][q]);
        x[ri] = __builtin_fmaf(-f, x[k], x[ri]);
      }
    }
    #pragma unroll
    for (int k = 4; k >= 0; --k) {
      float s = x[k];
      #pragma unroll
      for (int q = k + 1; q < 5; ++q) s = __builtin_fmaf(-A[k][q], x[q], s);
      x[k] = s / A[k][k];
    }
    #pragma unroll
    for (int p = 0; p < 5; ++p) {
      float cu = cur[(size_t)b * 5 + p];
      float d  = x[p];
      d = (d != d) ? (-0.1f * cu) : d;             // NaN -> -0.1*cur
      float nv = cu + d;
      nv = fmaxf(nv, prmm[2 * p]);
      nv = fminf(nv, prmm[2 * p + 1]);
      out[(size_t)b * 6 + p] = nv;
    }
    out[(size_t)b * 6 + 5] = red[20];
  }
}

extern "C" void kernel_launch(void* const* d_in, const int* in_sizes, int n_in,
                              void* d_out, int out_size, void* d_ws, size_t ws_size,
                              hipStream_t stream) {
  (void)n_in; (void)out_size; (void)d_ws; (void)ws_size;
  const float* smp  = (const float*)d_in[0];
  const float* mu   = (const float*)d_in[1];
  const float* jac  = (const float*)d_in[2];
  const float* cur  = (const float*)d_in[3];
  const float* prmm = (const float*)d_in[4];
  float* out = (float*)d_out;

  const int B = in_sizes[3] / PDIM;     // cur is [B,5]
  const int N = in_sizes[0] / B;        // K*H*W = 1536 (multiple of 64)
  dim3 grid((B + WAVES_PB - 1) / WAVES_PB);
  lm_mle_simflux_kernel<<<grid, THREADS_PB, 0, stream>>>(smp, mu, jac, cur, prmm, out, B, N);
}